// LSTMAutoencoder_9431748182279
// MI455X (gfx1250) — compile-verified
//
#include <hip/hip_runtime.h>

#define VOCAB 2000
#define EDIM  8
#define HDIM  16
#define BATCH 256
#define SLEN  128

typedef float v2f __attribute__((ext_vector_type(2)));
typedef float v8f __attribute__((ext_vector_type(8)));

__device__ __forceinline__ float sigm(float x) {
    // safe: exp(-x) -> inf gives 1/(1+inf)=0; -> 0 gives 1.
    return 1.0f / (1.0f + __expf(-x));
}
__device__ __forceinline__ float tanh_f(float x) {
    // overflow-safe tanh via exp of a non-positive argument
    float t = __expf(-2.0f * fabsf(x));
    float r = (1.0f - t) / (1.0f + t);
    return copysignf(r, x);
}

// ---------------------------------------------------------------------------
// Kernel 1: embedding gather + encoder scan + decoder scan -> hdec [B*S, H]
// One wave32 per batch element: barriers are S_NOP (single-wave WG), LDS ops
// are in-order within the wave. Lane j owns gates j and j+32 (weights in VGPRs).
// ---------------------------------------------------------------------------
__global__ void __launch_bounds__(32) lstm_kernel(
    const int*   __restrict__ seq,
    const float* __restrict__ emb,
    const float* __restrict__ eWih, const float* __restrict__ eWhh, const float* __restrict__ eb,
    const float* __restrict__ dWih, const float* __restrict__ dWhh, const float* __restrict__ db,
    float* __restrict__ hdec)
{
    __shared__ float xs[SLEN * EDIM];   // 4 KB: whole sequence's embeddings
    __shared__ float hbuf[HDIM];
    __shared__ float zbuf[4 * HDIM];

    const int b    = blockIdx.x;
    const int lane = threadIdx.x;
    const int g0   = lane;
    const int g1   = lane + 32;

    // Gather embeddings for this sequence into LDS (E=8 -> i>>3 / i&7).
    for (int i = lane; i < SLEN * EDIM; i += 32) {
        int t   = i >> 3;
        int e   = i & 7;
        int tok = seq[b * SLEN + t];
        xs[i] = emb[tok * EDIM + e];
    }
    if (lane < HDIM) hbuf[lane] = 0.0f;
    __syncthreads();

    // Per-lane weight rows (fully register-resident).
    float ewx0[EDIM], ewx1[EDIM], ewh0[HDIM], ewh1[HDIM];
    float dwx0[EDIM], dwx1[EDIM], dwh0[HDIM], dwh1[HDIM];
#pragma unroll
    for (int k = 0; k < EDIM; ++k) {
        ewx0[k] = eWih[g0 * EDIM + k];  ewx1[k] = eWih[g1 * EDIM + k];
        dwx0[k] = dWih[g0 * EDIM + k];  dwx1[k] = dWih[g1 * EDIM + k];
    }
#pragma unroll
    for (int k = 0; k < HDIM; ++k) {
        ewh0[k] = eWhh[g0 * HDIM + k];  ewh1[k] = eWhh[g1 * HDIM + k];
        dwh0[k] = dWhh[g0 * HDIM + k];  dwh1[k] = dWhh[g1 * HDIM + k];
    }
    const float ebb0 = eb[g0], ebb1 = eb[g1];
    const float dbb0 = db[g0], dbb1 = db[g1];

    float c_reg = 0.0f;   // lanes 0..15 hold c[lane]

    // -------- encoder scan (only final h,c survive) --------
    for (int t = 0; t < SLEN; ++t) {
        float z0 = ebb0, z1 = ebb1;
        const float* xt = &xs[t * EDIM];
#pragma unroll
        for (int k = 0; k < EDIM; ++k) { float x = xt[k];   z0 = fmaf(ewx0[k], x, z0); z1 = fmaf(ewx1[k], x, z1); }
#pragma unroll
        for (int k = 0; k < HDIM; ++k) { float h = hbuf[k]; z0 = fmaf(ewh0[k], h, z0); z1 = fmaf(ewh1[k], h, z1); }
        zbuf[g0] = z0; zbuf[g1] = z1;
        __syncthreads();
        if (lane < HDIM) {
            float iz = zbuf[lane], fz = zbuf[lane + 16], gz = zbuf[lane + 32], oz = zbuf[lane + 48];
            c_reg = sigm(fz) * c_reg + sigm(iz) * tanh_f(gz);
            hbuf[lane] = sigm(oz) * tanh_f(c_reg);
        }
        __syncthreads();
    }

    // -------- decoder scan (teacher-forced on same x), store all h --------
    for (int t = 0; t < SLEN; ++t) {
        float z0 = dbb0, z1 = dbb1;
        const float* xt = &xs[t * EDIM];
#pragma unroll
        for (int k = 0; k < EDIM; ++k) { float x = xt[k];   z0 = fmaf(dwx0[k], x, z0); z1 = fmaf(dwx1[k], x, z1); }
#pragma unroll
        for (int k = 0; k < HDIM; ++k) { float h = hbuf[k]; z0 = fmaf(dwh0[k], h, z0); z1 = fmaf(dwh1[k], h, z1); }
        zbuf[g0] = z0; zbuf[g1] = z1;
        __syncthreads();
        if (lane < HDIM) {
            float iz = zbuf[lane], fz = zbuf[lane + 16], gz = zbuf[lane + 32], oz = zbuf[lane + 48];
            c_reg = sigm(fz) * c_reg + sigm(iz) * tanh_f(gz);
            float h = sigm(oz) * tanh_f(c_reg);
            hbuf[lane] = h;
            hdec[(b * SLEN + t) * HDIM + lane] = h;   // [B*S, H] row-major
        }
        __syncthreads();
    }
}

// ---------------------------------------------------------------------------
// Kernel 2: logits[m, n] = hdec[m, :16] . Wout[n, :16] + bout[n]
// V_WMMA_F32_16X16X4_F32, K=16 as 4 chained K=4 steps. One wave per 16-row
// M-tile, looping over 25 N-tiles (grid.y = 5 chunks -> 125 tiles = 2000 cols).
// A/B fragments are built with per-lane selects only (no cross-lane traffic):
//   A 16x4 layout: v0 = lanes<16 ? K=4kk : K=4kk+2 ; v1 = K=4kk+1 : K=4kk+3
//   B  4x16 layout mirrors it; lane L holds column n = tile*16 + (L&15).
// Output stores are non-temporal (262 MB stream > 192 MB L2, never re-read).
// ---------------------------------------------------------------------------
__global__ void __launch_bounds__(256) proj_kernel(
    const float* __restrict__ hdec,
    const float* __restrict__ Wout,
    const float* __restrict__ bout,
    float* __restrict__ out)
{
    const int  lane  = threadIdx.x & 31;
    const int  wave  = threadIdx.x >> 5;
    const int  mtile = blockIdx.x * 8 + wave;       // 0..2047
    const bool hi    = lane >= 16;
    const int  l15   = lane & 15;

    // A-matrix: lane L holds hdec row (mtile*16 + (L&15)), 16 floats.
    const float4* arow = (const float4*)(hdec + (size_t)(mtile * 16 + l15) * HDIM);
    float4 a0 = arow[0], a1 = arow[1], a2 = arow[2], a3 = arow[3];
    v2f A0, A1, A2, A3;
    A0.x = hi ? a0.z : a0.x;  A0.y = hi ? a0.w : a0.y;
    A1.x = hi ? a1.z : a1.x;  A1.y = hi ? a1.w : a1.y;
    A2.x = hi ? a2.z : a2.x;  A2.y = hi ? a2.w : a2.y;
    A3.x = hi ? a3.z : a3.x;  A3.y = hi ? a3.w : a3.y;

    const int ntile0 = blockIdx.y * 25;
    for (int nt = 0; nt < 25; ++nt) {
        const int col = (ntile0 + nt) * 16 + l15;   // this lane's output column
        // B-matrix (= Wout^T tile): lane L holds Wout row 'col' (16 floats).
        const float4* wr = (const float4*)(Wout + (size_t)col * HDIM);
        float4 w0 = wr[0], w1 = wr[1], w2 = wr[2], w3 = wr[3];
        v2f B0, B1, B2, B3;
        B0.x = hi ? w0.z : w0.x;  B0.y = hi ? w0.w : w0.y;
        B1.x = hi ? w1.z : w1.x;  B1.y = hi ? w1.w : w1.y;
        B2.x = hi ? w2.z : w2.x;  B2.y = hi ? w2.w : w2.y;
        B3.x = hi ? w3.z : w3.x;  B3.y = hi ? w3.w : w3.y;

        // C starts as the bias (bout[col] is uniform across the lane's 8 rows).
        float bias = bout[col];
        v8f c;
#pragma unroll
        for (int i = 0; i < 8; ++i) c[i] = bias;

        // D = A*B + C, exact f32, K = 4+4+4+4
        c = __builtin_amdgcn_wmma_f32_16x16x4_f32(false, A0, false, B0, (short)0, c, false, false);
        c = __builtin_amdgcn_wmma_f32_16x16x4_f32(false, A1, false, B1, (short)0, c, false, false);
        c = __builtin_amdgcn_wmma_f32_16x16x4_f32(false, A2, false, B2, (short)0, c, false, false);
        c = __builtin_amdgcn_wmma_f32_16x16x4_f32(false, A3, false, B3, (short)0, c, false, false);

        // D layout: VGPR i -> row (i or i+8), col = lane&15 per half-wave.
        float* obase = out + (size_t)(mtile * 16 + (hi ? 8 : 0)) * VOCAB + col;
#pragma unroll
        for (int i = 0; i < 8; ++i)
            __builtin_nontemporal_store(c[i], obase + (size_t)i * VOCAB);
    }
}

extern "C" void kernel_launch(void* const* d_in, const int* in_sizes, int n_in,
                              void* d_out, int out_size, void* d_ws, size_t ws_size,
                              hipStream_t stream) {
    const int*   seq  = (const int*)d_in[0];
    const float* emb  = (const float*)d_in[1];
    const float* eWih = (const float*)d_in[2];
    const float* eWhh = (const float*)d_in[3];
    const float* ebv  = (const float*)d_in[4];
    const float* dWih = (const float*)d_in[5];
    const float* dWhh = (const float*)d_in[6];
    const float* dbv  = (const float*)d_in[7];
    const float* Wout = (const float*)d_in[8];
    const float* bv   = (const float*)d_in[9];
    float* out  = (float*)d_out;
    float* hdec = (float*)d_ws;   // B*S*H f32 = 2 MB scratch, fully overwritten

    lstm_kernel<<<BATCH, 32, 0, stream>>>(seq, emb, eWih, eWhh, ebv,
                                          dWih, dWhh, dbv, hdec);

    // 2048 M-tiles (8 waves/block * 256 blocks) x 5 N-chunks of 25 tiles.
    proj_kernel<<<dim3((BATCH * SLEN) / 128, 5), 256, 0, stream>>>(hdec, Wout, bv, out);
}